// GCN_82085414961434
// MI455X (gfx1250) — compile-verified
//
#include <hip/hip_runtime.h>
#include <hip/hip_bf16.h>

#define D 128
#define EPS 1e-5f

typedef __attribute__((ext_vector_type(2))) float v2f;
typedef __attribute__((ext_vector_type(8))) float v8f;

// ---------------- utility ----------------
__global__ void k_zero(float* __restrict__ p, long n) {
    long i = (long)blockIdx.x * blockDim.x + threadIdx.x;
    if (i < n) p[i] = 0.0f;
}

// x[i,f] = atom_emb[x_idx[i], f]
__global__ void k_gather(const int* __restrict__ xi, const float* __restrict__ emb,
                         float* __restrict__ x, int n) {
    long i = (long)blockIdx.x * blockDim.x + threadIdx.x;
    if (i < (long)n * D) {
        int node = (int)(i >> 7);
        int f = (int)(i & (D - 1));
        x[i] = emb[(long)xi[node] * D + f];
    }
}

// deg[row[e]] += 1
__global__ void k_degcnt(const int* __restrict__ row, float* __restrict__ deg, int e) {
    int i = blockIdx.x * blockDim.x + threadIdx.x;
    if (i < e) atomicAdd(&deg[row[i]], 1.0f);
}

// deg += 1 (self loop), dinv = deg^-1/2
__global__ void k_degfin(float* __restrict__ deg, float* __restrict__ dinv, int n) {
    int i = blockIdx.x * blockDim.x + threadIdx.x;
    if (i < n) {
        float d = deg[i] + 1.0f;
        deg[i] = d;
        dinv[i] = rsqrtf(d);
    }
}

// ---------------- WMMA GEMM: h = x @ W^T + b ----------------
// One wave -> 16 rows x 128 cols. fp32 exact via V_WMMA_F32_16X16X4_F32.
__global__ __launch_bounds__(256) void k_gemm(const float* __restrict__ x,
                                              const float* __restrict__ W,
                                              const float* __restrict__ bias,
                                              float* __restrict__ h, int n) {
    int lane = threadIdx.x & 31;
    int wave = threadIdx.x >> 5;
    int m0 = blockIdx.x * 128 + wave * 16;
    if (m0 >= n) return;  // wave-uniform: EXEC stays all-ones for WMMA

    int lm = lane & 15;
    int kh = (lane >> 4) << 1;   // 0 or 2 : K-half within the 16x4 A tile
    int mh = (lane >> 4) << 3;   // 0 or 8 : M-half for C/D store

    int row = m0 + lm;
    if (row >= n) row = n - 1;   // clamp (stores are masked instead)
    const float* xr = x + (long)row * D;

    v8f acc[8];
#pragma unroll
    for (int nb = 0; nb < 8; ++nb) {
        float bv = bias[nb * 16 + lm];   // bias[n], same for every M
#pragma unroll
        for (int r = 0; r < 8; ++r) acc[nb][r] = bv;
    }

#pragma unroll 4
    for (int kb = 0; kb < 32; ++kb) {
        int k = kb * 4 + kh;
        v2f a;
        a.x = xr[k];
        a.y = xr[k + 1];
#pragma unroll
        for (int nb = 0; nb < 8; ++nb) {
            // B[k][n] = W[n][k]  (h = x @ W^T)
            const float* wr = W + (long)(nb * 16 + lm) * D + k;
            v2f b;
            b.x = wr[0];
            b.y = wr[1];
            acc[nb] = __builtin_amdgcn_wmma_f32_16x16x4_f32(
                false, a, false, b, (short)0, acc[nb], false, false);
        }
    }

#pragma unroll
    for (int nb = 0; nb < 8; ++nb) {
        int col = nb * 16 + lm;
#pragma unroll
        for (int r = 0; r < 8; ++r) {
            int m = m0 + mh + r;   // VGPR r: M = r (lanes<16) / r+8 (lanes>=16)
            if (m < n) h[(long)m * D + col] = acc[nb][r];
        }
    }
}

// ---------------- edge message + scatter: one wave per edge ----------------
__global__ __launch_bounds__(256) void k_edge(const int* __restrict__ row,
                                              const int* __restrict__ col,
                                              const int* __restrict__ eattr,
                                              const float* __restrict__ h,
                                              const float* __restrict__ bond,
                                              const float* __restrict__ dinv,
                                              float* __restrict__ agg, int e) {
    int w = (int)(((long)blockIdx.x * blockDim.x + threadIdx.x) >> 5);
    int lane = threadIdx.x & 31;
    if (w >= e) return;
    int r = row[w];
    int c = col[w];
    int a = eattr[w];
    float nrm = dinv[r] * dinv[c];
    float4 hv = ((const float4*)(h + (long)r * D))[lane];
    float4 bv = ((const float4*)(bond + (long)a * D))[lane];
    float4 m;
    m.x = fmaxf(hv.x + bv.x, 0.0f) * nrm;
    m.y = fmaxf(hv.y + bv.y, 0.0f) * nrm;
    m.z = fmaxf(hv.z + bv.z, 0.0f) * nrm;
    m.w = fmaxf(hv.w + bv.w, 0.0f) * nrm;
    float* ap = agg + (long)c * D + lane * 4;
    atomicAdd(ap + 0, m.x);
    atomicAdd(ap + 1, m.y);
    atomicAdd(ap + 2, m.z);
    atomicAdd(ap + 3, m.w);
}

// ---------------- conv = agg + relu(h+root)/deg ; BN partial sums ----------------
__global__ __launch_bounds__(256) void k_conv_stats(float* __restrict__ conv,
                                                    const float* __restrict__ h,
                                                    const float* __restrict__ root,
                                                    const float* __restrict__ deg,
                                                    float* __restrict__ sum,
                                                    float* __restrict__ sumsq, int n) {
    __shared__ float sb[256];
    __shared__ float sq[256];
    int f = threadIdx.x & (D - 1);
    int half = threadIdx.x >> 7;
    float rv = root[f];
    float s = 0.0f, ss = 0.0f;
    for (int i = blockIdx.x * 2 + half; i < n; i += gridDim.x * 2) {
        long idx = (long)i * D + f;
        float cv = conv[idx] + fmaxf(h[idx] + rv, 0.0f) / deg[i];
        conv[idx] = cv;
        s += cv;
        ss += cv * cv;
    }
    sb[threadIdx.x] = s;
    sq[threadIdx.x] = ss;
    __syncthreads();
    if (threadIdx.x < D) {
        atomicAdd(&sum[f], sb[threadIdx.x] + sb[threadIdx.x + 128]);
        atomicAdd(&sumsq[f], sq[threadIdx.x] + sq[threadIdx.x + 128]);
    }
}

__global__ void k_bnfin(const float* __restrict__ sum, const float* __restrict__ sumsq,
                        float* __restrict__ mu, float* __restrict__ inv, int n) {
    int f = threadIdx.x;
    if (f < D) {
        float m = sum[f] / (float)n;
        float v = sumsq[f] / (float)n - m * m;   // biased var
        mu[f] = m;
        inv[f] = rsqrtf(v + EPS);
    }
}

// x += relu((conv - mu) * inv * gamma + beta)
__global__ void k_update(float* __restrict__ x, const float* __restrict__ conv,
                         const float* __restrict__ mu, const float* __restrict__ inv,
                         const float* __restrict__ gamma, const float* __restrict__ beta,
                         int n) {
    long i = (long)blockIdx.x * blockDim.x + threadIdx.x;
    if (i < (long)n * D) {
        int f = (int)(i & (D - 1));
        float bn = (conv[i] - mu[f]) * inv[f] * gamma[f] + beta[f];
        x[i] += fmaxf(bn, 0.0f);
    }
}

// ---------------- final edge prediction: one wave per edge ----------------
__global__ __launch_bounds__(256) void k_pred(const float* __restrict__ x,
                                              const int* __restrict__ row,
                                              const int* __restrict__ col,
                                              const float* __restrict__ pw,
                                              const float* __restrict__ pb,
                                              float* __restrict__ out, int e) {
    int w = (int)(((long)blockIdx.x * blockDim.x + threadIdx.x) >> 5);
    int lane = threadIdx.x & 31;
    if (w >= e) return;
    int r = row[w];
    int c = col[w];
    float4 xr = ((const float4*)(x + (long)r * D))[lane];
    float4 xc = ((const float4*)(x + (long)c * D))[lane];
    float4 w0 = ((const float4*)pw)[lane];
    float4 w1 = ((const float4*)(pw + D))[lane];
    float p = xr.x * w0.x + xr.y * w0.y + xr.z * w0.z + xr.w * w0.w +
              xc.x * w1.x + xc.y * w1.y + xc.z * w1.z + xc.w * w1.w;
#pragma unroll
    for (int off = 16; off > 0; off >>= 1) p += __shfl_xor(p, off, 32);
    if (lane == 0) out[w] = p + pb[0];
}

// ---------------- host launcher ----------------
extern "C" void kernel_launch(void* const* d_in, const int* in_sizes, int n_in,
                              void* d_out, int out_size, void* d_ws, size_t ws_size,
                              hipStream_t stream) {
    const int*   x_idx = (const int*)d_in[0];
    const int*   eidx  = (const int*)d_in[1];
    const int*   eattr = (const int*)d_in[2];
    const float* emb   = (const float*)d_in[3];
    const float* linW  = (const float*)d_in[4];
    const float* linb  = (const float*)d_in[5];
    const float* root  = (const float*)d_in[6];
    const float* bond  = (const float*)d_in[7];
    const float* gamma = (const float*)d_in[8];
    const float* beta  = (const float*)d_in[9];
    const float* predW = (const float*)d_in[10];
    const float* predb = (const float*)d_in[11];
    float* out = (float*)d_out;

    const int N = in_sizes[0];
    const int E = in_sizes[2];
    const int L = in_sizes[5] / D;   // linb is [L, D]
    const int* row = eidx;
    const int* col = eidx + E;

    float* ws    = (float*)d_ws;
    float* x     = ws;                      // N*D
    float* h     = x + (size_t)N * D;       // N*D
    float* agg   = h + (size_t)N * D;       // N*D
    float* deg   = agg + (size_t)N * D;     // N
    float* dinv  = deg + N;                 // N
    float* sum   = dinv + N;                // 128
    float* sumsq = sum + D;                 // 128
    float* mu    = sumsq + D;               // 128
    float* inv   = mu + D;                  // 128

    auto cdiv = [](long a, long b) { return (int)((a + b - 1) / b); };
    const long ND = (long)N * D;

    k_zero<<<cdiv(N, 256), 256, 0, stream>>>(deg, N);
    k_gather<<<cdiv(ND, 256), 256, 0, stream>>>(x_idx, emb, x, N);
    k_degcnt<<<cdiv(E, 256), 256, 0, stream>>>(row, deg, E);
    k_degfin<<<cdiv(N, 256), 256, 0, stream>>>(deg, dinv, N);

    for (int l = 0; l < L; ++l) {
        k_gemm<<<cdiv(N, 128), 256, 0, stream>>>(x, linW + (size_t)l * D * D,
                                                 linb + (size_t)l * D, h, N);
        k_zero<<<cdiv(ND, 256), 256, 0, stream>>>(agg, ND);
        k_zero<<<1, 256, 0, stream>>>(sum, 2 * D);  // sum + sumsq contiguous
        k_edge<<<cdiv((long)E * 32, 256), 256, 0, stream>>>(
            row, col, eattr, h, bond + (size_t)l * 4 * D, dinv, agg, E);
        k_conv_stats<<<512, 256, 0, stream>>>(agg, h, root + (size_t)l * D, deg,
                                              sum, sumsq, N);
        k_bnfin<<<1, 128, 0, stream>>>(sum, sumsq, mu, inv, N);
        k_update<<<cdiv(ND, 256), 256, 0, stream>>>(x, agg, mu, inv,
                                                    gamma + (size_t)l * D,
                                                    beta + (size_t)l * D, N);
    }

    k_pred<<<cdiv((long)E * 32, 256), 256, 0, stream>>>(x, row, col, predW, predb,
                                                        out, E);
}